// Model_6373731467274
// MI455X (gfx1250) — compile-verified
//
#include <hip/hip_runtime.h>
#include <cstddef>

// ---------------------------------------------------------------------------
// Problem constants (from reference: B,T,D,H,K = 32,512,256,512,8)
// ---------------------------------------------------------------------------
#define B_ 32
#define T_ 512
#define D_ 256
#define H_ 512
#define K_ 8

constexpr float LOG2PI_ = 1.8378770664093453f;   // ln(2*pi)
constexpr float LOGVAR_ = -9.210340371976184f;   // ln(1e-4)
constexpr float INVVAR_ = 10000.0f;              // 1/VAR

// Output tuple layout (flat, in reference return order)
constexpr size_t N_PI = (size_t)B_ * K_;              // prob_initial
constexpr size_t N_PS = (size_t)B_ * T_ * K_ * K_;    // prob_all_s
constexpr size_t N_PH = (size_t)B_ * T_ * K_;         // prob_all_h
constexpr size_t N_PY = (size_t)B_ * T_;              // prob_all_y
constexpr size_t N_G1 = N_PH;                         // gamma1
constexpr size_t N_G2 = (size_t)B_ * (T_ - 1) * K_ * K_; // gamma2
constexpr size_t N_FW = N_PH;                         // forward_prob
constexpr size_t O_PI = 0;
constexpr size_t O_PS = O_PI + N_PI;
constexpr size_t O_PH = O_PS + N_PS;
constexpr size_t O_PY = O_PH + N_PH;
constexpr size_t O_G1 = O_PY + N_PY;
constexpr size_t O_G2 = O_G1 + N_G1;
constexpr size_t O_FW = O_G2 + N_G2;
constexpr size_t O_BW = O_FW + N_FW;

// ---------------------------------------------------------------------------
// WMMA f32 16x16x4 (full-precision f32 matrix core path on gfx1250)
// A: 16x4, per lane: M = lane%16, K = (lane>=16 ? 2 : 0) + {0,1}
// B: 4x16, per lane: N = lane%16, K = (lane>=16 ? 2 : 0) + {0,1}
// C/D: 8 VGPRs, elem i: M = i + (lane>=16 ? 8 : 0), N = lane%16
// ---------------------------------------------------------------------------
typedef __attribute__((ext_vector_type(2))) float v2f;
typedef __attribute__((ext_vector_type(8))) float v8f;

__device__ inline v8f wmma4(v2f a, v2f b, v8f c) {
  return __builtin_amdgcn_wmma_f32_16x16x4_f32(
      false, a, false, b, (short)0, c, false, false);
}

// ---------------------------------------------------------------------------
// Init: prob_initial, prob_all_s[:,0]=eye, prob_all_h[:,0]=lp0
// ---------------------------------------------------------------------------
__global__ __launch_bounds__(256) void k_init(const float* __restrict__ ini,
                                              float* __restrict__ pi,
                                              float* __restrict__ ps,
                                              float* __restrict__ ph) {
  const int idx = blockIdx.x * 256 + threadIdx.x;  // 0..2047
  if (idx < B_ * 64) {
    const int b = idx >> 6, i = (idx >> 3) & 7, j = idx & 7;
    ps[(((size_t)b * T_) * 8 + i) * 8 + j] = (i == j) ? 1.f : 0.f;
  }
  if (idx < B_ * 8) {
    const int b = idx >> 3, k = idx & 7;
    float mx = ini[0];
    for (int q = 1; q < 8; q++) mx = fmaxf(mx, ini[q]);
    float s = 0.f;
    for (int q = 0; q < 8; q++) s += expf(ini[q] - mx);
    pi[(size_t)b * 8 + k] = ini[k] - (mx + logf(s));
    ph[((size_t)b * T_) * 8 + k] = -0.5f * (float)H_ * (LOG2PI_ + LOGVAR_);
  }
}

// ---------------------------------------------------------------------------
// Big fused kernel: pre = x1·Wih^T + hprev·Whh^T + bih + bhh
// then tanh, (sampled_h - tanh)^2, reduce over H  ->  prob_all_h[:,1:,k]
// M-tile = 32 rows (two A fragments share each B fragment: 8 WMMA / 4 B-loads)
// grid = (511 row tiles, 8 k), block = 256 (8 waves, 64 cols each)
// ---------------------------------------------------------------------------
__global__ __launch_bounds__(256) void k_pre(
    const float* __restrict__ x, const float* __restrict__ sh,
    const float* __restrict__ Wih, const float* __restrict__ Whh,
    const float* __restrict__ bih, const float* __restrict__ bhh,
    float* __restrict__ ph) {
  const int mt = blockIdx.x, k = blockIdx.y;
  const int tid = threadIdx.x;
  const int wave = tid >> 5, lane = tid & 31;
  const int lo = lane & 15, hi = lane >> 4;
  const int row0 = mt * 32;

  // Two A-fragment rows (two 16-row halves of the 32-row tile)
  const int arow0 = row0 + lo;
  const int arow1 = row0 + 16 + lo;
  const int ab0 = arow0 / (T_ - 1), at0 = arow0 % (T_ - 1);
  const int ab1 = arow1 / (T_ - 1), at1 = arow1 % (T_ - 1);
  const float* xrow0 = x + ((size_t)ab0 * T_ + at0 + 1) * D_;   // x[:,1:]
  const float* hrow0 = sh + ((size_t)ab0 * T_ + at0) * H_;      // h_prev
  const float* xrow1 = x + ((size_t)ab1 * T_ + at1 + 1) * D_;
  const float* hrow1 = sh + ((size_t)ab1 * T_ + at1) * H_;

  const int cbase = wave * 64;  // this wave's first h column (4 tiles of 16)

  v8f acc[2][4];
  v8f z = {};
#pragma unroll
  for (int hf = 0; hf < 2; hf++)
#pragma unroll
    for (int ct = 0; ct < 4; ct++) acc[hf][ct] = z;

  // ---- reduction over D (x · Wih) ----
  const float* wp[4];
#pragma unroll
  for (int ct = 0; ct < 4; ct++)
    wp[ct] = Wih + ((size_t)k * H_ + cbase + ct * 16 + lo) * D_;
  for (int r = 0; r < D_; r += 4) {
    v2f a0, a1;
    a0.x = xrow0[r + hi * 2];
    a0.y = xrow0[r + hi * 2 + 1];
    a1.x = xrow1[r + hi * 2];
    a1.y = xrow1[r + hi * 2 + 1];
#pragma unroll
    for (int ct = 0; ct < 4; ct++) {
      v2f b;
      b.x = wp[ct][r + hi * 2];
      b.y = wp[ct][r + hi * 2 + 1];
      acc[0][ct] = wmma4(a0, b, acc[0][ct]);
      acc[1][ct] = wmma4(a1, b, acc[1][ct]);
    }
  }
  // ---- reduction over H (h_prev · Whh) ----
  const float* up[4];
#pragma unroll
  for (int ct = 0; ct < 4; ct++)
    up[ct] = Whh + ((size_t)k * H_ + cbase + ct * 16 + lo) * H_;
  for (int r = 0; r < H_; r += 4) {
    v2f a0, a1;
    a0.x = hrow0[r + hi * 2];
    a0.y = hrow0[r + hi * 2 + 1];
    a1.x = hrow1[r + hi * 2];
    a1.y = hrow1[r + hi * 2 + 1];
#pragma unroll
    for (int ct = 0; ct < 4; ct++) {
      v2f b;
      b.x = up[ct][r + hi * 2];
      b.y = up[ct][r + hi * 2 + 1];
      acc[0][ct] = wmma4(a0, b, acc[0][ct]);
      acc[1][ct] = wmma4(a1, b, acc[1][ct]);
    }
  }

  // ---- epilogue: bias, tanh, squared error vs sampled_h[:,1:], H-reduction ----
  float part[2][8];
#pragma unroll
  for (int hf = 0; hf < 2; hf++)
#pragma unroll
    for (int i = 0; i < 8; i++) part[hf][i] = 0.f;
#pragma unroll
  for (int ct = 0; ct < 4; ct++) {
    const int h = cbase + ct * 16 + lo;
    const float bias = bih[k * H_ + h] + bhh[k * H_ + h];
#pragma unroll
    for (int hf = 0; hf < 2; hf++) {
#pragma unroll
      for (int i = 0; i < 8; i++) {
        const int grow = row0 + hf * 16 + i + hi * 8;
        const int gb = grow / (T_ - 1), gt = grow % (T_ - 1);
        const float tv = tanhf(acc[hf][ct][i] + bias);
        const float dlt = sh[((size_t)gb * T_ + gt + 1) * H_ + h] - tv;
        part[hf][i] += dlt * dlt;
      }
    }
  }

  __shared__ float red[32][128];
  __shared__ float red2[32][16];
#pragma unroll
  for (int hf = 0; hf < 2; hf++)
#pragma unroll
    for (int i = 0; i < 8; i++)
      red[hf * 16 + i + hi * 8][wave * 16 + lo] = part[hf][i];
  __syncthreads();
#pragma unroll
  for (int hf = 0; hf < 2; hf++) {
    const int m = hf * 16 + (tid >> 4), j = tid & 15;
    float s = 0.f;
#pragma unroll
    for (int w = 0; w < 8; w++) s += red[m][w * 16 + j];
    red2[m][j] = s;
  }
  __syncthreads();
  if (tid < 32) {
    float s = 0.f;
#pragma unroll
    for (int j = 0; j < 16; j++) s += red2[tid][j];
    const int grow = row0 + tid;
    const int gb = grow / (T_ - 1), gt = grow % (T_ - 1);
    ph[((size_t)gb * T_ + gt + 1) * K_ + k] =
        -0.5f * ((float)H_ * (LOG2PI_ + LOGVAR_) + s * INVVAR_);
  }
}

// ---------------------------------------------------------------------------
// trans = tanh(h_prev · Wt_ih^T + bt), normalize (lse over i) -> prob_all_s[:,1:]
// grid = 1022, block = 128 (4 waves, 16 cols each)
// ---------------------------------------------------------------------------
__global__ __launch_bounds__(128) void k_trans(const float* __restrict__ sh,
                                               const float* __restrict__ Wt,
                                               const float* __restrict__ btv,
                                               float* __restrict__ ps) {
  const int mt = blockIdx.x;
  const int tid = threadIdx.x;
  const int wave = tid >> 5, lane = tid & 31, lo = lane & 15, hi = lane >> 4;
  const int row0 = mt * 16;
  const int arow = row0 + lo;
  const int ab = arow / (T_ - 1), at = arow % (T_ - 1);
  const float* hrow = sh + ((size_t)ab * T_ + at) * H_;
  const int c = wave * 16 + lo;  // column 0..63
  const float* wr = Wt + (size_t)c * H_;

  v8f acc = {};
  for (int r = 0; r < H_; r += 4) {
    v2f a;
    a.x = hrow[r + hi * 2];
    a.y = hrow[r + hi * 2 + 1];
    v2f b;
    b.x = wr[r + hi * 2];
    b.y = wr[r + hi * 2 + 1];
    acc = wmma4(a, b, acc);
  }

  __shared__ float tr[16][64];
  const float bias = btv[c];
#pragma unroll
  for (int i = 0; i < 8; i++) tr[i + hi * 8][c] = tanhf(acc[i] + bias);
  __syncthreads();

  // normalize over destination state i (axis 2 of (B,T-1,K,K))
  const int m = tid >> 3, j = tid & 7;
  float mx = tr[m][j];
#pragma unroll
  for (int i = 1; i < 8; i++) mx = fmaxf(mx, tr[m][i * 8 + j]);
  float s = 0.f;
#pragma unroll
  for (int i = 0; i < 8; i++) s += expf(tr[m][i * 8 + j] - mx);
  const float l = mx + logf(s);
  const int grow = row0 + m;
  const int gb = grow / (T_ - 1), gt = grow % (T_ - 1);
  float* outp = ps + (((size_t)gb * T_ + gt + 1) * 8) * 8 + j;
#pragma unroll
  for (int i = 0; i < 8; i++) outp[i * 8] = tr[m][i * 8 + j] - l;
}

// ---------------------------------------------------------------------------
// MLP (relu(sh·W1^T) -> relu(·W2^T) -> ·W3^T) fused with gauss-lp vs y_train
// grid = 1024 (16 rows each), block = 256 (8 waves)
// ---------------------------------------------------------------------------
__global__ __launch_bounds__(256) void k_mlp(
    const float* __restrict__ sh, const float* __restrict__ y,
    const float* __restrict__ W1, const float* __restrict__ b1,
    const float* __restrict__ W2, const float* __restrict__ b2,
    const float* __restrict__ W3, const float* __restrict__ b3,
    float* __restrict__ py) {
  const int mt = blockIdx.x;
  const int tid = threadIdx.x;
  const int wave = tid >> 5, lane = tid & 31, lo = lane & 15, hi = lane >> 4;
  const int row0 = mt * 16;

  __shared__ float h1s[16][32];
  __shared__ float h2s[16][64];
  __shared__ float red[16][128];
  __shared__ float red2[16][16];

  // stage 1: h1 = relu(sh @ W1^T + b1)   (16x32, red 512) -- waves 0..1
  if (wave < 2) {
    const int c = wave * 16 + lo;
    const float* arow = sh + (size_t)(row0 + lo) * H_;
    const float* wr = W1 + (size_t)c * H_;
    v8f acc = {};
    for (int r = 0; r < H_; r += 4) {
      v2f a;
      a.x = arow[r + hi * 2];
      a.y = arow[r + hi * 2 + 1];
      v2f b;
      b.x = wr[r + hi * 2];
      b.y = wr[r + hi * 2 + 1];
      acc = wmma4(a, b, acc);
    }
    const float bias = b1[c];
#pragma unroll
    for (int i = 0; i < 8; i++) h1s[i + hi * 8][c] = fmaxf(acc[i] + bias, 0.f);
  }
  __syncthreads();

  // stage 2: h2 = relu(h1 @ W2^T + b2)   (16x64, red 32) -- waves 0..3
  if (wave < 4) {
    const int c = wave * 16 + lo;
    const float* wr = W2 + (size_t)c * 32;
    v8f acc = {};
#pragma unroll
    for (int r = 0; r < 32; r += 4) {
      v2f a;
      a.x = h1s[lo][r + hi * 2];
      a.y = h1s[lo][r + hi * 2 + 1];
      v2f b;
      b.x = wr[r + hi * 2];
      b.y = wr[r + hi * 2 + 1];
      acc = wmma4(a, b, acc);
    }
    const float bias = b2[c];
#pragma unroll
    for (int i = 0; i < 8; i++) h2s[i + hi * 8][c] = fmaxf(acc[i] + bias, 0.f);
  }
  __syncthreads();

  // stage 3: em = h2 @ W3^T + b3  (16x256, red 64) + gauss lp over D
  float part[8];
#pragma unroll
  for (int i = 0; i < 8; i++) part[i] = 0.f;
#pragma unroll
  for (int ct = 0; ct < 2; ct++) {
    const int c = wave * 32 + ct * 16 + lo;  // 0..255
    const float* wr = W3 + (size_t)c * 64;
    v8f acc = {};
#pragma unroll
    for (int r = 0; r < 64; r += 4) {
      v2f a;
      a.x = h2s[lo][r + hi * 2];
      a.y = h2s[lo][r + hi * 2 + 1];
      v2f b;
      b.x = wr[r + hi * 2];
      b.y = wr[r + hi * 2 + 1];
      acc = wmma4(a, b, acc);
    }
    const float bias = b3[c];
#pragma unroll
    for (int i = 0; i < 8; i++) {
      const int grow = row0 + i + hi * 8;
      const float em = acc[i] + bias;
      const float dlt = y[(size_t)grow * D_ + c] - em;
      part[i] += dlt * dlt;
    }
  }
#pragma unroll
  for (int i = 0; i < 8; i++) red[i + hi * 8][wave * 16 + lo] = part[i];
  __syncthreads();
  {
    const int m = tid >> 4, j = tid & 15;
    float s = 0.f;
#pragma unroll
    for (int w = 0; w < 8; w++) s += red[m][w * 16 + j];
    red2[m][j] = s;
  }
  __syncthreads();
  if (tid < 16) {
    float s = 0.f;
#pragma unroll
    for (int j = 0; j < 16; j++) s += red2[tid][j];
    py[row0 + tid] = -0.5f * ((float)D_ * (LOG2PI_ + LOGVAR_) + s * INVVAR_);
  }
}

// ---------------------------------------------------------------------------
// Forward scan: one block per batch, 64 threads = (i,j) pairs, T-1 steps
// ---------------------------------------------------------------------------
__global__ __launch_bounds__(64) void k_forward(const float* __restrict__ pi,
                                                const float* __restrict__ ps,
                                                const float* __restrict__ ph,
                                                const float* __restrict__ py,
                                                float* __restrict__ fw) {
  const int b = blockIdx.x;
  const int tid = threadIdx.x;
  const int i = tid >> 3, j = tid & 7;
  __shared__ float alpha[8];
  __shared__ float vbuf[64];
  __shared__ float anew[8];
  if (tid < 8) {
    const float a0 = pi[b * 8 + tid] + ph[((size_t)b * T_) * 8 + tid] + py[(size_t)b * T_];
    alpha[tid] = a0;
    fw[((size_t)b * T_) * 8 + tid] = a0;
  }
  __syncthreads();
  for (int t = 1; t < T_; t++) {
    vbuf[tid] = alpha[j] + ps[(((size_t)b * T_ + t) * 8 + i) * 8 + j];
    __syncthreads();
    if (j == 0) {
      float mx = vbuf[i * 8];
#pragma unroll
      for (int q = 1; q < 8; q++) mx = fmaxf(mx, vbuf[i * 8 + q]);
      float s = 0.f;
#pragma unroll
      for (int q = 0; q < 8; q++) s += expf(vbuf[i * 8 + q] - mx);
      anew[i] = ph[((size_t)b * T_ + t) * 8 + i] + py[(size_t)b * T_ + t] + mx + logf(s);
    }
    __syncthreads();
    if (tid < 8) {
      alpha[tid] = anew[tid];
      fw[((size_t)b * T_ + t) * 8 + tid] = anew[tid];
    }
    __syncthreads();
  }
}

// ---------------------------------------------------------------------------
// Backward scan
// ---------------------------------------------------------------------------
__global__ __launch_bounds__(64) void k_backward(const float* __restrict__ ps,
                                                 const float* __restrict__ ph,
                                                 const float* __restrict__ py,
                                                 float* __restrict__ bw) {
  const int b = blockIdx.x;
  const int tid = threadIdx.x;
  const int i = tid >> 3, j = tid & 7;
  __shared__ float beta[8];
  __shared__ float vbuf[64];
  __shared__ float bnew[8];
  if (tid < 8) {
    beta[tid] = 0.f;
    bw[((size_t)b * T_ + (T_ - 1)) * 8 + tid] = 0.f;
  }
  __syncthreads();
  for (int t = T_ - 2; t >= 0; t--) {
    const float e = ph[((size_t)b * T_ + t + 1) * 8 + i] + py[(size_t)b * T_ + t + 1];
    vbuf[tid] = beta[i] + e + ps[(((size_t)b * T_ + t + 1) * 8 + i) * 8 + j];
    __syncthreads();
    if (i == 0) {  // tid == j here
      float mx = vbuf[j];
#pragma unroll
      for (int q = 1; q < 8; q++) mx = fmaxf(mx, vbuf[q * 8 + j]);
      float s = 0.f;
#pragma unroll
      for (int q = 0; q < 8; q++) s += expf(vbuf[q * 8 + j] - mx);
      bnew[j] = mx + logf(s);
    }
    __syncthreads();
    if (tid < 8) {
      beta[tid] = bnew[tid];
      bw[((size_t)b * T_ + t) * 8 + tid] = bnew[tid];
    }
    __syncthreads();
  }
}

// ---------------------------------------------------------------------------
// gamma1 = (fw+bw) - lse over states
// ---------------------------------------------------------------------------
__global__ __launch_bounds__(256) void k_gamma1(const float* __restrict__ fw,
                                                const float* __restrict__ bw,
                                                float* __restrict__ g1) {
  const int row = blockIdx.x * 256 + threadIdx.x;  // 0..B*T-1
  if (row >= B_ * T_) return;
  float v[8];
  float mx = -3.4e38f;
#pragma unroll
  for (int q = 0; q < 8; q++) {
    v[q] = fw[(size_t)row * 8 + q] + bw[(size_t)row * 8 + q];
    mx = fmaxf(mx, v[q]);
  }
  float s = 0.f;
#pragma unroll
  for (int q = 0; q < 8; q++) s += expf(v[q] - mx);
  const float l = mx + logf(s);
#pragma unroll
  for (int q = 0; q < 8; q++) g1[(size_t)row * 8 + q] = v[q] - l;
}

// ---------------------------------------------------------------------------
// gamma2: pair lse over (i,j); one block per (b,t), 64 threads
// ---------------------------------------------------------------------------
__global__ __launch_bounds__(64) void k_gamma2(const float* __restrict__ fw,
                                               const float* __restrict__ bw,
                                               const float* __restrict__ ps,
                                               const float* __restrict__ ph,
                                               const float* __restrict__ py,
                                               float* __restrict__ g2) {
  const int bt = blockIdx.x;  // 0..B*(T-1)-1
  const int b = bt / (T_ - 1), t = bt % (T_ - 1);
  const int tid = threadIdx.x, i = tid >> 3, j = tid & 7;
  __shared__ float vbuf[64];
  __shared__ float lse_s;
  const float pair = fw[((size_t)b * T_ + t) * 8 + j] +
                     ps[(((size_t)b * T_ + t + 1) * 8 + i) * 8 + j] +
                     ph[((size_t)b * T_ + t + 1) * 8 + i] +
                     py[(size_t)b * T_ + t + 1] +
                     bw[((size_t)b * T_ + t + 1) * 8 + i];
  vbuf[tid] = pair;
  __syncthreads();
  if (tid == 0) {
    float mx = vbuf[0];
    for (int q = 1; q < 64; q++) mx = fmaxf(mx, vbuf[q]);
    float s = 0.f;
    for (int q = 0; q < 64; q++) s += expf(vbuf[q] - mx);
    lse_s = mx + logf(s);
  }
  __syncthreads();
  g2[((size_t)bt * 8 + i) * 8 + j] = pair - lse_s;
}

// ---------------------------------------------------------------------------
extern "C" void kernel_launch(void* const* d_in, const int* in_sizes, int n_in,
                              void* d_out, int out_size, void* d_ws, size_t ws_size,
                              hipStream_t stream) {
  const float* x    = (const float*)d_in[0];
  const float* y    = (const float*)d_in[1];
  const float* sh   = (const float*)d_in[2];
  const float* ini  = (const float*)d_in[3];
  const float* Wih  = (const float*)d_in[4];
  const float* Whh  = (const float*)d_in[5];
  const float* bih  = (const float*)d_in[6];
  const float* bhh  = (const float*)d_in[7];
  const float* Wt   = (const float*)d_in[8];
  const float* btv  = (const float*)d_in[9];
  const float* W1   = (const float*)d_in[10];
  const float* b1   = (const float*)d_in[11];
  const float* W2   = (const float*)d_in[12];
  const float* b2   = (const float*)d_in[13];
  const float* W3   = (const float*)d_in[14];
  const float* b3   = (const float*)d_in[15];

  float* out = (float*)d_out;
  float* pi = out + O_PI;
  float* ps = out + O_PS;
  float* ph = out + O_PH;
  float* py = out + O_PY;
  float* g1 = out + O_G1;
  float* g2 = out + O_G2;
  float* fw = out + O_FW;
  float* bw = out + O_BW;

  k_init<<<8, 256, 0, stream>>>(ini, pi, ps, ph);
  k_pre<<<dim3((B_ * (T_ - 1)) / 32, K_), 256, 0, stream>>>(x, sh, Wih, Whh, bih, bhh, ph);
  k_trans<<<(B_ * (T_ - 1)) / 16, 128, 0, stream>>>(sh, Wt, btv, ps);
  k_mlp<<<(B_ * T_) / 16, 256, 0, stream>>>(sh, y, W1, b1, W2, b2, W3, b3, py);
  k_forward<<<B_, 64, 0, stream>>>(pi, ps, ph, py, fw);
  k_backward<<<B_, 64, 0, stream>>>(ps, ph, py, bw);
  k_gamma1<<<(B_ * T_ + 255) / 256, 256, 0, stream>>>(fw, bw, g1);
  k_gamma2<<<B_ * (T_ - 1), 64, 0, stream>>>(fw, bw, ps, ph, py, g2);
}